// STGCN_33268816675400
// MI455X (gfx1250) — compile-verified
//
#include <hip/hip_runtime.h>
#include <math.h>

#define DYN 16
#define IN 32
#define IN_ES 36
#define IN_ET 20
#define SCORE_D 51

typedef float v2f __attribute__((ext_vector_type(2)));
typedef float v8f __attribute__((ext_vector_type(8)));

__device__ __forceinline__ float gelu_f(float x) {
    const float c = 0.7978845608028654f; // sqrt(2/pi)
    float x3 = x * x * x;
    return 0.5f * x * (1.0f + tanhf(c * (x + 0.044715f * x3)));
}
__device__ __forceinline__ float sigmoid_f(float x) { return 1.0f / (1.0f + expf(-x)); }
__device__ __forceinline__ float silu_f(float x) { return x * sigmoid_f(x); }

// One wave computes O[16][32] = X[16][K] @ W[K][32] with V_WMMA_F32_16X16X4_F32.
// Xl: LDS, 16 rows, row stride ldx. Wl: LDS [K][32]. Ol: LDS [16][32]. K % 4 == 0.
// A layout (32-bit A 16x4): lane L holds row M=L&15, K = v + 2*(L>>4) in VGPR v.
// B layout mirrored: lane L holds col N=L&15, K = v + 2*(L>>4).
// D layout: VGPR v, lane L -> row v + 8*(L>>4), col L&15.
__device__ __forceinline__ void wave_gemm_16x32(const float* Xl, int ldx,
                                                const float* Wl, float* Ol,
                                                int K, int lane) {
    v8f acc0 = {0.f,0.f,0.f,0.f,0.f,0.f,0.f,0.f};
    v8f acc1 = {0.f,0.f,0.f,0.f,0.f,0.f,0.f,0.f};
    const int m  = lane & 15;
    const int kg = lane >> 4; // 0 or 1
    for (int k0 = 0; k0 < K; k0 += 4) {
        const int ka = k0 + 2 * kg;
        v2f a;  a.x  = Xl[m * ldx + ka];          a.y  = Xl[m * ldx + ka + 1];
        v2f b0; b0.x = Wl[ka * 32 + m];           b0.y = Wl[(ka + 1) * 32 + m];
        v2f b1; b1.x = Wl[ka * 32 + 16 + m];      b1.y = Wl[(ka + 1) * 32 + 16 + m];
        acc0 = __builtin_amdgcn_wmma_f32_16x16x4_f32(false, a, false, b0, (short)0, acc0, false, false);
        acc1 = __builtin_amdgcn_wmma_f32_16x16x4_f32(false, a, false, b1, (short)0, acc1, false, false);
    }
#pragma unroll
    for (int v = 0; v < 8; ++v) {
        int row = v + 8 * kg;
        Ol[row * 32 + m]      = acc0[v];
        Ol[row * 32 + 16 + m] = acc1[v];
    }
}

// ---------------- Kernel 1: per-node aux (time-edge MLP, scores, control update) ----
__global__ __launch_bounds__(256) void node_aux_kernel(
    const float* __restrict__ h, const float* __restrict__ h_dyn,
    const float* __restrict__ t_feat, const float* __restrict__ w_time_e,
    const int* __restrict__ src_t,
    const float* __restrict__ te_w1, const float* __restrict__ te_b1,
    const float* __restrict__ te_g,  const float* __restrict__ te_be,
    const float* __restrict__ te_w2, const float* __restrict__ te_b2,
    const float* __restrict__ ss_w,  const float* __restrict__ st_w,
    const float* __restrict__ cu_w1, const float* __restrict__ cu_b1,
    const float* __restrict__ cu_w2, const float* __restrict__ cu_b2,
    const float* __restrict__ cu_g,  const float* __restrict__ cu_be,
    const float* __restrict__ cu_w3, const float* __restrict__ cu_b3,
    float* __restrict__ w_t_out, float* __restrict__ score_s_out,
    float* __restrict__ score_t_out, float* __restrict__ h_dyn_new, int n)
{
    __shared__ float sTeW1[IN_ET * 32];
    __shared__ float sCuW1[SCORE_D * 32];
    __shared__ float sCuW2[32 * 32];
    __shared__ float sCuW3[32 * DYN];
    __shared__ float sTeB1[32], sTeG[32], sTeBe[32], sTeW2[32];
    __shared__ float sCuB1[32], sCuB2[32], sCuG[32], sCuBe[32], sCuB3[DYN];
    __shared__ float sSs[SCORE_D], sSt[SCORE_D];

    int tid = threadIdx.x;
    for (int i = tid; i < IN_ET * 32; i += 256)    sTeW1[i] = te_w1[i];
    for (int i = tid; i < SCORE_D * 32; i += 256)  sCuW1[i] = cu_w1[i];
    for (int i = tid; i < 32 * 32; i += 256)       sCuW2[i] = cu_w2[i];
    for (int i = tid; i < 32 * DYN; i += 256)      sCuW3[i] = cu_w3[i];
    if (tid < 32) {
        sTeB1[tid] = te_b1[tid]; sTeG[tid] = te_g[tid]; sTeBe[tid] = te_be[tid]; sTeW2[tid] = te_w2[tid];
        sCuB1[tid] = cu_b1[tid]; sCuB2[tid] = cu_b2[tid]; sCuG[tid] = cu_g[tid]; sCuBe[tid] = cu_be[tid];
    }
    if (tid < DYN) sCuB3[tid] = cu_b3[tid];
    if (tid < SCORE_D) { sSs[tid] = ss_w[tid]; sSt[tid] = st_w[tid]; }
    __syncthreads();

    int i = blockIdx.x * 256 + tid;
    if (i >= n) return;
    float t0 = t_feat[0], t1 = t_feat[1], t2 = t_feat[2];

    // ---- time-edge MLP: [w_ii, t, h_dyn[src_t]] -> w_t
    float f[IN_ET];
    f[0] = w_time_e[i]; f[1] = t0; f[2] = t1; f[3] = t2;
    int sidx = src_t[i];
    {
        const float4* hv = (const float4*)(h_dyn + (size_t)sidx * DYN); // 64B-aligned rows
#pragma unroll
        for (int k = 0; k < 4; ++k) {
            float4 v = hv[k];
            f[4 + 4*k] = v.x; f[5 + 4*k] = v.y; f[6 + 4*k] = v.z; f[7 + 4*k] = v.w;
        }
    }
    float zt[32]; float mean = 0.f;
#pragma unroll
    for (int j = 0; j < 32; ++j) {
        float acc = sTeB1[j];
        for (int k = 0; k < IN_ET; ++k) acc += f[k] * sTeW1[k * 32 + j];
        acc = gelu_f(acc); zt[j] = acc; mean += acc;
    }
    mean *= (1.f / 32.f);
    float var = 0.f;
#pragma unroll
    for (int j = 0; j < 32; ++j) { float d = zt[j] - mean; var += d * d; }
    var *= (1.f / 32.f);
    float rinv = rsqrtf(var + 1e-5f);
    float wt = te_b2[0];
#pragma unroll
    for (int j = 0; j < 32; ++j) wt += ((zt[j] - mean) * rinv * sTeG[j] + sTeBe[j]) * sTeW2[j];
    w_t_out[i] = wt;

    // ---- z = [h, h_dyn, t]  (51)
    float z[SCORE_D];
    {
        const float4* hv = (const float4*)(h + (size_t)i * IN);  // 128B-aligned rows
#pragma unroll
        for (int k = 0; k < 8; ++k) {
            float4 v = hv[k];
            z[4*k] = v.x; z[4*k+1] = v.y; z[4*k+2] = v.z; z[4*k+3] = v.w;
        }
        const float4* dv = (const float4*)(h_dyn + (size_t)i * DYN);
#pragma unroll
        for (int k = 0; k < 4; ++k) {
            float4 v = dv[k];
            z[IN+4*k] = v.x; z[IN+4*k+1] = v.y; z[IN+4*k+2] = v.z; z[IN+4*k+3] = v.w;
        }
    }
    z[48] = t0; z[49] = t1; z[50] = t2;
    float as = 0.f, at = 0.f;
#pragma unroll
    for (int k = 0; k < SCORE_D; ++k) { as += z[k] * sSs[k]; at += z[k] * sSt[k]; }
    score_s_out[i] = sigmoid_f(as);
    score_t_out[i] = sigmoid_f(at);

    // ---- control update: silu -> linear -> LN -> silu -> linear
    float c[32];
#pragma unroll
    for (int j = 0; j < 32; ++j) {
        float acc = sCuB1[j];
        for (int k = 0; k < SCORE_D; ++k) acc += z[k] * sCuW1[k * 32 + j];
        c[j] = silu_f(acc);
    }
    float c2[32]; mean = 0.f;
#pragma unroll
    for (int j = 0; j < 32; ++j) {
        float acc = sCuB2[j];
        for (int k = 0; k < 32; ++k) acc += c[k] * sCuW2[k * 32 + j];
        c2[j] = acc; mean += acc;
    }
    mean *= (1.f / 32.f); var = 0.f;
#pragma unroll
    for (int j = 0; j < 32; ++j) { float d = c2[j] - mean; var += d * d; }
    var *= (1.f / 32.f); rinv = rsqrtf(var + 1e-5f);
#pragma unroll
    for (int j = 0; j < 32; ++j) c2[j] = silu_f((c2[j] - mean) * rinv * sCuG[j] + sCuBe[j]);
#pragma unroll
    for (int k = 0; k < DYN; ++k) {
        float acc = sCuB3[k];
        for (int j = 0; j < 32; ++j) acc += c2[j] * sCuW3[j * DYN + k];
        h_dyn_new[(size_t)i * DYN + k] = h_dyn[(size_t)i * DYN + k] + acc;
    }
}

// ---------------- Kernel 2: spatial edge MLP via WMMA (128 edges / block) ------------
__global__ __launch_bounds__(256) void edge_space_kernel(
    const float* __restrict__ w_space_e, const float* __restrict__ t_feat,
    const float* __restrict__ h_dyn, const int* __restrict__ src_s,
    const int* __restrict__ dst_s,
    const float* __restrict__ se_w1, const float* __restrict__ se_b1,
    const float* __restrict__ se_g,  const float* __restrict__ se_be,
    const float* __restrict__ se_w2, const float* __restrict__ se_b2,
    float* __restrict__ w_s_out, int E)
{
    __shared__ float sW1[IN_ES * 32];
    __shared__ __align__(16) float sX[128 * IN_ES];  // row stride 144B (16B multiple)
    __shared__ float sH[128 * 32];
    __shared__ float sB1[32], sG[32], sBe[32], sW2[32];

    int tid = threadIdx.x;
    for (int i = tid; i < IN_ES * 32; i += 256) sW1[i] = se_w1[i];
    if (tid < 32) { sB1[tid] = se_b1[tid]; sG[tid] = se_g[tid]; sBe[tid] = se_be[tid]; sW2[tid] = se_w2[tid]; }

    int e0 = blockIdx.x * 128;
    if (tid < 128) {
        int e = e0 + tid;
        float* r = &sX[tid * IN_ES];
        if (e < E) {
            r[0] = w_space_e[e];
            r[1] = t_feat[0]; r[2] = t_feat[1]; r[3] = t_feat[2];
            int s = src_s[e], d = dst_s[e];
            const float4* hs = (const float4*)(h_dyn + (size_t)s * DYN); // 64B-aligned
            const float4* hd = (const float4*)(h_dyn + (size_t)d * DYN);
            float4* rs = (float4*)(r + 4);   // +16B
            float4* rd = (float4*)(r + 20);  // +80B
#pragma unroll
            for (int k = 0; k < 4; ++k) { rs[k] = hs[k]; rd[k] = hd[k]; }
        } else {
            for (int k = 0; k < IN_ES; ++k) r[k] = 0.f;
        }
    }
    __syncthreads();
    int wave = tid >> 5, lane = tid & 31;
    wave_gemm_16x32(&sX[wave * 16 * IN_ES], IN_ES, sW1, &sH[wave * 16 * 32], IN_ES, lane);
    __syncthreads();
    if (tid < 128) {
        int e = e0 + tid;
        if (e < E) {
            float* hr = &sH[tid * 32];
            float mean = 0.f;
#pragma unroll
            for (int j = 0; j < 32; ++j) { float x = gelu_f(hr[j] + sB1[j]); hr[j] = x; mean += x; }
            mean *= (1.f / 32.f);
            float var = 0.f;
#pragma unroll
            for (int j = 0; j < 32; ++j) { float d = hr[j] - mean; var += d * d; }
            var *= (1.f / 32.f);
            float rinv = rsqrtf(var + 1e-5f);
            float acc = se_b2[0];
#pragma unroll
            for (int j = 0; j < 32; ++j) acc += ((hr[j] - mean) * rinv * sG[j] + sBe[j]) * sW2[j];
            w_s_out[e] = acc;
        }
    }
}

// ---------------- Kernel 3: zero accumulators ---------------------------------------
__global__ void zero_kernel(float* __restrict__ p, long long nelem) {
    long long i = (long long)blockIdx.x * blockDim.x + threadIdx.x;
    if (i < nelem) p[i] = 0.f;
}

// ---------------- Kernel 4: weighted scatter-add message passing --------------------
// One thread per (edge, column): each wave's 32 lanes cover one contiguous 128B row,
// so both the gather load and the atomic add coalesce into single transactions.
__global__ __launch_bounds__(256) void scatter_kernel(
    const float* __restrict__ feat, const float* __restrict__ w,
    const int* __restrict__ src_idx, const int* __restrict__ dst_idx,
    float* __restrict__ accum, int num)
{
    long long t = (long long)blockIdx.x * 256 + threadIdx.x;
    int e = (int)(t >> 5);
    int j = (int)(t & 31);
    if (e >= num) return;
    float we = w[e];
    int s = src_idx[e], d = dst_idx[e];
    atomicAdd(&accum[(size_t)d * 32 + j], feat[(size_t)s * 32 + j] * we);
}

// ---------------- Kernel 5: node update MLP via 3x WMMA GEMMs (64 nodes / block) ----
__global__ __launch_bounds__(128) void node_update_kernel(
    const float* __restrict__ h, const float* __restrict__ dh_space,
    const float* __restrict__ dh_time, const float* __restrict__ h_dyn_new,
    const float* __restrict__ score_s, const float* __restrict__ score_t,
    const float* __restrict__ t_feat,
    const float* __restrict__ fn_w1, const float* __restrict__ fn_b1,
    const float* __restrict__ fn_w2, const float* __restrict__ fn_b2,
    const float* __restrict__ fn_g,  const float* __restrict__ fn_be,
    const float* __restrict__ fn_w3, const float* __restrict__ fn_b3,
    float* __restrict__ out, int n)
{
    const int KP = 116; // 115 padded to multiple of 4; row stride 464B (16B multiple)
    __shared__ __align__(16) float sU[64 * 116];   // u panel; later reused for sH2/sH3
    __shared__ float sW1[116 * 32];
    __shared__ float sH1[64 * 32];
    __shared__ float sW2[32 * 32];
    __shared__ float sW3[32 * 32];
    __shared__ float sB1[32], sB2[32], sG[32], sBe[32], sB3[32];

    int tid = threadIdx.x;
    for (int idx = tid; idx < 116 * 32; idx += 128) sW1[idx] = (idx < 115 * 32) ? fn_w1[idx] : 0.f;
    for (int idx = tid; idx < 32 * 32; idx += 128) { sW2[idx] = fn_w2[idx]; sW3[idx] = fn_w3[idx]; }
    if (tid < 32) { sB1[tid] = fn_b1[tid]; sB2[tid] = fn_b2[tid]; sG[tid] = fn_g[tid]; sBe[tid] = fn_be[tid]; sB3[tid] = fn_b3[tid]; }

    int i0 = blockIdx.x * 64;
    if (tid < 64) {
        int i = i0 + tid;
        float* r = &sU[tid * KP];
        if (i < n) {
            float ss = score_s[i], stt = score_t[i];
            const float4* hv = (const float4*)(h        + (size_t)i * 32);
            const float4* sv = (const float4*)(dh_space + (size_t)i * 32);
            const float4* tv = (const float4*)(dh_time  + (size_t)i * 32);
            const float4* dv = (const float4*)(h_dyn_new+ (size_t)i * DYN);
            float4* r0 = (float4*)r;          // +0B
            float4* r1 = (float4*)(r + 32);   // +128B
            float4* r2 = (float4*)(r + 64);   // +256B
            float4* r3 = (float4*)(r + 96);   // +384B
#pragma unroll
            for (int k = 0; k < 8; ++k) r0[k] = hv[k];
#pragma unroll
            for (int k = 0; k < 8; ++k) {
                float4 v = sv[k];
                v.x *= ss; v.y *= ss; v.z *= ss; v.w *= ss;
                r1[k] = v;
            }
#pragma unroll
            for (int k = 0; k < 8; ++k) {
                float4 v = tv[k];
                v.x *= stt; v.y *= stt; v.z *= stt; v.w *= stt;
                r2[k] = v;
            }
#pragma unroll
            for (int k = 0; k < 4; ++k) r3[k] = dv[k];
            r[112] = t_feat[0]; r[113] = t_feat[1]; r[114] = t_feat[2]; r[115] = 0.f;
        } else {
            for (int k = 0; k < KP; ++k) r[k] = 0.f;
        }
    }
    __syncthreads();
    int wave = tid >> 5, lane = tid & 31;
    // GEMM1: u(115) -> 32
    wave_gemm_16x32(&sU[wave * 16 * KP], KP, sW1, &sH1[wave * 16 * 32], KP, lane);
    __syncthreads();
    if (tid < 64) {
        float* hr = &sH1[tid * 32];
#pragma unroll
        for (int j = 0; j < 32; ++j) hr[j] = gelu_f(hr[j] + sB1[j]);
    }
    __syncthreads();
    float* sH2 = sU;             // u panel no longer needed
    float* sH3 = sU + 64 * 32;
    // GEMM2: 32 -> 32
    wave_gemm_16x32(&sH1[wave * 16 * 32], 32, sW2, &sH2[wave * 16 * 32], 32, lane);
    __syncthreads();
    if (tid < 64) {
        float* r = &sH2[tid * 32];
        float mean = 0.f;
#pragma unroll
        for (int j = 0; j < 32; ++j) { float x = r[j] + sB2[j]; r[j] = x; mean += x; }
        mean *= (1.f / 32.f);
        float var = 0.f;
#pragma unroll
        for (int j = 0; j < 32; ++j) { float d = r[j] - mean; var += d * d; }
        var *= (1.f / 32.f);
        float rinv = rsqrtf(var + 1e-5f);
#pragma unroll
        for (int j = 0; j < 32; ++j) r[j] = gelu_f((r[j] - mean) * rinv * sG[j] + sBe[j]);
    }
    __syncthreads();
    // GEMM3: 32 -> 32
    wave_gemm_16x32(&sH2[wave * 16 * 32], 32, sW3, &sH3[wave * 16 * 32], 32, lane);
    __syncthreads();
    if (tid < 64) {
        int i = i0 + tid;
        if (i < n) {
#pragma unroll
            for (int j = 0; j < 32; ++j) out[(size_t)i * 32 + j] = sH3[tid * 32 + j] + sB3[j];
        }
    }
}

extern "C" void kernel_launch(void* const* d_in, const int* in_sizes, int n_in,
                              void* d_out, int out_size, void* d_ws, size_t ws_size,
                              hipStream_t stream)
{
    const float* h         = (const float*)d_in[0];
    const float* h_prev    = (const float*)d_in[1];
    const float* h_dyn     = (const float*)d_in[2];
    const float* t_feat    = (const float*)d_in[3];
    const float* w_space_e = (const float*)d_in[4];
    const float* w_time_e  = (const float*)d_in[5];
    const int*   src_s     = (const int*)d_in[6];
    const int*   dst_s     = (const int*)d_in[7];
    const int*   src_t     = (const int*)d_in[8];
    const int*   dst_t     = (const int*)d_in[9];
    const float* se_w1 = (const float*)d_in[10]; const float* se_b1 = (const float*)d_in[11];
    const float* se_g  = (const float*)d_in[12]; const float* se_be = (const float*)d_in[13];
    const float* se_w2 = (const float*)d_in[14]; const float* se_b2 = (const float*)d_in[15];
    const float* te_w1 = (const float*)d_in[16]; const float* te_b1 = (const float*)d_in[17];
    const float* te_g  = (const float*)d_in[18]; const float* te_be = (const float*)d_in[19];
    const float* te_w2 = (const float*)d_in[20]; const float* te_b2 = (const float*)d_in[21];
    const float* ss_w  = (const float*)d_in[22]; const float* st_w  = (const float*)d_in[23];
    const float* cu_w1 = (const float*)d_in[24]; const float* cu_b1 = (const float*)d_in[25];
    const float* cu_w2 = (const float*)d_in[26]; const float* cu_b2 = (const float*)d_in[27];
    const float* cu_g  = (const float*)d_in[28]; const float* cu_be = (const float*)d_in[29];
    const float* cu_w3 = (const float*)d_in[30]; const float* cu_b3 = (const float*)d_in[31];
    const float* fn_w1 = (const float*)d_in[32]; const float* fn_b1 = (const float*)d_in[33];
    const float* fn_w2 = (const float*)d_in[34]; const float* fn_b2 = (const float*)d_in[35];
    const float* fn_g  = (const float*)d_in[36]; const float* fn_be = (const float*)d_in[37];
    const float* fn_w3 = (const float*)d_in[38]; const float* fn_b3 = (const float*)d_in[39];

    int E = in_sizes[4];   // w_space_e has E elements
    int n = in_sizes[5];   // w_time_e has N elements

    // workspace layout (floats)
    float* ws        = (float*)d_ws;
    float* dh_space  = ws;                                   // n*32
    float* dh_time   = dh_space + (size_t)n * 32;            // n*32
    float* w_s       = dh_time  + (size_t)n * 32;            // E
    float* w_t       = w_s + (size_t)E;                      // n
    float* score_s   = w_t + n;                              // n
    float* score_t   = score_s + n;                          // n
    float* h_dyn_new = score_t + n;                          // n*16

    node_aux_kernel<<<(n + 255) / 256, 256, 0, stream>>>(
        h, h_dyn, t_feat, w_time_e, src_t,
        te_w1, te_b1, te_g, te_be, te_w2, te_b2, ss_w, st_w,
        cu_w1, cu_b1, cu_w2, cu_b2, cu_g, cu_be, cu_w3, cu_b3,
        w_t, score_s, score_t, h_dyn_new, n);

    edge_space_kernel<<<(E + 127) / 128, 256, 0, stream>>>(
        w_space_e, t_feat, h_dyn, src_s, dst_s,
        se_w1, se_b1, se_g, se_be, se_w2, se_b2, w_s, E);

    long long zn = (long long)n * 64;
    zero_kernel<<<(int)((zn + 255) / 256), 256, 0, stream>>>(dh_space, zn);

    // (edge, column) mapping: num*32 threads
    long long ts = (long long)E * 32;
    scatter_kernel<<<(int)((ts + 255) / 256), 256, 0, stream>>>(h, w_s, src_s, dst_s, dh_space, E);
    long long tt = (long long)n * 32;
    scatter_kernel<<<(int)((tt + 255) / 256), 256, 0, stream>>>(h_prev, w_t, src_t, dst_t, dh_time, n);

    node_update_kernel<<<(n + 63) / 64, 128, 0, stream>>>(
        h, dh_space, dh_time, h_dyn_new, score_s, score_t, t_feat,
        fn_w1, fn_b1, fn_w2, fn_b2, fn_g, fn_be, fn_w3, fn_b3,
        (float*)d_out, n);
}